// SGFE_9199819948248
// MI455X (gfx1250) — compile-verified
//
#include <hip/hip_runtime.h>
#include <stdint.h>

// ---------------------------------------------------------------------------
// SGFE pipeline for MI455X (gfx1250), all GEMMs via V_WMMA_F32_16X16X4_F32.
// f32 WMMA chosen deliberately: total ~20 GFLOP vs ~400MB traffic + ~100M
// atomics -> memory/atomic bound; f32 keeps reference precision for free.
// ---------------------------------------------------------------------------

typedef __attribute__((ext_vector_type(2))) float v2f;
typedef __attribute__((ext_vector_type(4))) float v4f;
typedef __attribute__((ext_vector_type(8))) float v8f;

static __device__ __forceinline__ v8f wmma4(v2f a, v2f b, v8f c) {
  // 8 args: (neg_a, A, neg_b, B, c_mod, C, reuse_a, reuse_b)
  return __builtin_amdgcn_wmma_f32_16x16x4_f32(false, a, false, b, (short)0, c,
                                               false, false);
}

// A fragment (16x4 f32) from a row-major tile (leading dim ld, ld even).
// lane l: M = l&15, Kpair = l>>4 -> elements [M][k0+2*Kpair .. +1]
static __device__ __forceinline__ v2f load_a_frag(const float* base, int ld,
                                                  int k0, int lane) {
  int m = lane & 15, kg = lane >> 4;
  return *(const v2f*)(base + m * ld + k0 + 2 * kg);
}

// B fragment (4x16 f32) from row-major global weights W[K x ldw].
// lane l: N = l&15, rows k0+2*(l>>4) and +1.
static __device__ __forceinline__ v2f load_b_frag(const float* __restrict__ W,
                                                  int ldw, int k0, int n0,
                                                  int lane) {
  int n = lane & 15, kg = lane >> 4;
  const float* p = W + (size_t)(k0 + 2 * kg) * ldw + n0 + n;
  v2f b;
  b.x = p[0];
  b.y = p[ldw];
  return b;
}

static __device__ __forceinline__ void atomic_max_f32(float* addr, float val) {
  // Native CDNA5 float max atomic (non-returning form).
  asm volatile("global_atomic_max_num_f32 %0, %1, off" ::"v"(
                   (uint64_t)(uintptr_t)addr),
               "v"(val)
               : "memory");
}

static __device__ __forceinline__ float sigmoidf_(float x) {
  return 1.0f / (1.0f + __expf(-x));
}

// ---------------------------------------------------------------------------
// Generic GEMM: C[M, NT*16] = act(rowscale(A[M,K]) @ W[K,NT*16] + bias)
// Block = 256 threads = 8 waves; each wave computes a 16-row strip; the block
// stages 128 rows of A in LDS (coalesced) incl. optional 1/max(cnt,1) scaling.
// ---------------------------------------------------------------------------
template <int NT, int K>
__global__ __launch_bounds__(256) void gemm_wmma_kernel(
    const float* __restrict__ A, const float* __restrict__ rowscale,
    const float* __restrict__ W, const float* __restrict__ bias,
    float* __restrict__ C, int M, int do_relu) {
  constexpr int LDA = K + 2;  // even -> 8B-aligned v2f fragment reads
  __shared__ float As[128 * LDA];

  const int tid = threadIdx.x;
  const int row0 = blockIdx.x * 128;

  // Stage A tile (128 x K) into LDS with float4 loads.
  for (int i = tid; i < 128 * (K / 4); i += 256) {
    int r = i / (K / 4);
    int c4 = (i % (K / 4)) * 4;
    int gr = row0 + r;
    gr = gr < M ? gr : M - 1;
    v4f v = *(const v4f*)(A + (size_t)gr * K + c4);
    float s = 1.0f;
    if (rowscale) s = 1.0f / fmaxf(rowscale[gr], 1.0f);
    float* dst = As + r * LDA + c4;
    dst[0] = v.x * s;
    dst[1] = v.y * s;
    dst[2] = v.z * s;
    dst[3] = v.w * s;
  }
  __syncthreads();

  const int wave = tid >> 5, lane = tid & 31;
  const float* Aw = As + (wave * 16) * LDA;

  v8f acc[NT];
  const v8f vzero = {0.f, 0.f, 0.f, 0.f, 0.f, 0.f, 0.f, 0.f};
#pragma unroll
  for (int t = 0; t < NT; ++t) acc[t] = vzero;

#pragma unroll
  for (int k0 = 0; k0 < K; k0 += 4) {
    v2f a = load_a_frag(Aw, LDA, k0, lane);
#pragma unroll
    for (int t = 0; t < NT; ++t) {
      v2f b = load_b_frag(W, NT * 16, k0, t * 16, lane);
      acc[t] = wmma4(a, b, acc[t]);
    }
  }

  // Epilogue: C/D layout -> lane = N + 16*(M>=8), vgpr = M&7.
  const int n = lane & 15, half = lane >> 4;
#pragma unroll
  for (int t = 0; t < NT; ++t) {
    float bv = bias ? bias[t * 16 + n] : 0.0f;
#pragma unroll
    for (int v = 0; v < 8; ++v) {
      int m = v + 8 * half;
      int gr = row0 + wave * 16 + m;
      if (gr < M) {
        float x = acc[t][v] + bv;
        if (do_relu) x = fmaxf(x, 0.0f);
        C[(size_t)gr * (NT * 16) + t * 16 + n] = x;
      }
    }
  }
}

// ---------------------------------------------------------------------------
// Fill / segment-sum / scatter-max helpers
// ---------------------------------------------------------------------------
__global__ void fill_kernel(float* __restrict__ p, long n, float v) {
  long i = (long)blockIdx.x * blockDim.x + threadIdx.x;
  if (i < n) p[i] = v;
}

__global__ void segsum_kernel(const float* __restrict__ red,
                              const int* __restrict__ inv,
                              float* __restrict__ ssum,
                              float* __restrict__ cnt, int N) {
  int t = blockIdx.x * blockDim.x + threadIdx.x;
  int vox = t >> 4;  // 16 threads/voxel, 4 channels each
  if (vox >= N) return;
  int c4 = (t & 15) * 4;
  int s = inv[vox];
  v4f val = *(const v4f*)(red + (size_t)vox * 64 + c4);
  float* dst = ssum + (size_t)s * 64 + c4;
  atomicAdd(dst + 0, val.x);
  atomicAdd(dst + 1, val.y);
  atomicAdd(dst + 2, val.z);
  atomicAdd(dst + 3, val.w);
  if (c4 == 0) atomicAdd(cnt + s, 1.0f);
}

__global__ void scatter_max_kernel(const float* __restrict__ proj,
                                   const int* __restrict__ inv_pts,
                                   const int* __restrict__ inv_out,
                                   float* __restrict__ out, int P) {
  int t = blockIdx.x * blockDim.x + threadIdx.x;
  int p = t >> 4;
  if (p >= P) return;
  int c4 = (t & 15) * 4;
  int v = inv_pts[p];
  int o = inv_out[p];
  v4f val = *(const v4f*)(proj + (size_t)v * 64 + c4);
  float* dst = out + (size_t)o * 64 + c4;
  atomic_max_f32(dst + 0, val.x);
  atomic_max_f32(dst + 1, val.y);
  atomic_max_f32(dst + 2, val.z);
  atomic_max_f32(dst + 3, val.w);
}

// ---------------------------------------------------------------------------
// Fused per-voxel tail: gather 4 scale feats, Z = relu(FS@W_fc),
// att_j = sigmoid(Z@W_fcs[j]+b), G = sum_j SF_j*att_j, FUSED = G@W_out,
// OUT = [red|FUSED], H = relu(OUT@W_lin1), P = H@W_lin2 + b -> proj.
// Block = 128 threads = 4 waves; each wave owns 16 rows + private LDS scratch.
// Per-wave LDS floats: SF 4*16*34=2176 (aliased by OUT 16*130=2080),
// FSG 16*34=544 (FS then G), ZH 16*66=1056 (Z then H). Total 3776*4 waves.
// ---------------------------------------------------------------------------
__global__ __launch_bounds__(128) void fuse_kernel(
    const float* __restrict__ red, const float* __restrict__ a0,
    const float* __restrict__ a1, const float* __restrict__ a2,
    const float* __restrict__ a3, const int* __restrict__ inv0,
    const int* __restrict__ inv1, const int* __restrict__ inv2,
    const int* __restrict__ inv3, const float* __restrict__ W_fc,
    const float* __restrict__ W_fcs, const float* __restrict__ b_fcs,
    const float* __restrict__ W_out, const float* __restrict__ W_lin1,
    const float* __restrict__ W_lin2, const float* __restrict__ b_lin2,
    float* __restrict__ proj, int M) {
  __shared__ float smem[4 * 3776];  // 60416 B

  const int tid = threadIdx.x;
  const int wave = tid >> 5, lane = tid & 31;
  const int rw0 = blockIdx.x * 64 + wave * 16;  // first row of this wave
  float* SF = smem + wave * 3776;               // [4][16][34], later OUT[16][130]
  float* FSG = SF + 2176;                       // [16][34]  (FS then G)
  float* ZH = FSG + 544;                        // [16][66]  (Z then H)
  float* OUT = SF;                              // alias (SF dead by then)

  const int n = lane & 15, half = lane >> 4;
  const v8f vzero = {0.f, 0.f, 0.f, 0.f, 0.f, 0.f, 0.f, 0.f};
  const int* invs[4] = {inv0, inv1, inv2, inv3};
  const float* atts[4] = {a0, a1, a2, a3};

  // Phase 0: gather SF[j][r][0:32] and accumulate FS = sum_j SF[j].
#pragma unroll
  for (int rq = 0; rq < 4; ++rq) {
    int r = rq * 4 + (lane >> 3);  // 4 rows per iter, 8 lanes/row
    int c4 = (lane & 7) * 4;
    int gr = rw0 + r;
    gr = gr < M ? gr : M - 1;
    v4f fs = {0.f, 0.f, 0.f, 0.f};
#pragma unroll
    for (int j = 0; j < 4; ++j) {
      int idx = invs[j][gr];
      v4f v = *(const v4f*)(atts[j] + (size_t)idx * 32 + c4);
      float* s = SF + (j * 16 + r) * 34 + c4;
      s[0] = v.x; s[1] = v.y; s[2] = v.z; s[3] = v.w;
      fs += v;
    }
    float* f = FSG + r * 34 + c4;
    f[0] = fs.x; f[1] = fs.y; f[2] = fs.z; f[3] = fs.w;
  }
  __syncthreads();

  // Phase 1: Z = relu(FS @ W_fc)  (16x32, K=32)
  {
    v8f cz[2] = {vzero, vzero};
#pragma unroll
    for (int k0 = 0; k0 < 32; k0 += 4) {
      v2f a = load_a_frag(FSG, 34, k0, lane);
      cz[0] = wmma4(a, load_b_frag(W_fc, 32, k0, 0, lane), cz[0]);
      cz[1] = wmma4(a, load_b_frag(W_fc, 32, k0, 16, lane), cz[1]);
    }
#pragma unroll
    for (int t = 0; t < 2; ++t)
#pragma unroll
      for (int v = 0; v < 8; ++v) {
        int m = v + 8 * half;
        ZH[m * 66 + t * 16 + n] = fmaxf(cz[t][v], 0.0f);
      }
  }
  __syncthreads();

  // Phase 2: G = sum_j SF[j] * sigmoid(Z @ W_fcs[j] + b_fcs[j])
  {
    v8f g[2] = {vzero, vzero};
#pragma unroll
    for (int j = 0; j < 4; ++j) {
      const float* Wj = W_fcs + j * 1024;
      v8f ca[2] = {vzero, vzero};
#pragma unroll
      for (int k0 = 0; k0 < 32; k0 += 4) {
        v2f a = load_a_frag(ZH, 66, k0, lane);
        ca[0] = wmma4(a, load_b_frag(Wj, 32, k0, 0, lane), ca[0]);
        ca[1] = wmma4(a, load_b_frag(Wj, 32, k0, 16, lane), ca[1]);
      }
#pragma unroll
      for (int t = 0; t < 2; ++t) {
        float bv = b_fcs[j * 32 + t * 16 + n];
#pragma unroll
        for (int v = 0; v < 8; ++v) {
          int m = v + 8 * half;
          float att = sigmoidf_(ca[t][v] + bv);
          g[t][v] += att * SF[(j * 16 + m) * 34 + t * 16 + n];
        }
      }
    }
    __syncthreads();
#pragma unroll
    for (int t = 0; t < 2; ++t)
#pragma unroll
      for (int v = 0; v < 8; ++v) {
        int m = v + 8 * half;
        FSG[m * 34 + t * 16 + n] = g[t][v];  // overwrite FS with G
      }
  }
  __syncthreads();

  // Phase 3: FUSED = G @ W_out (16x64, K=32); OUT = [red | FUSED]
  {
    v8f cf[4] = {vzero, vzero, vzero, vzero};
#pragma unroll
    for (int k0 = 0; k0 < 32; k0 += 4) {
      v2f a = load_a_frag(FSG, 34, k0, lane);
#pragma unroll
      for (int t = 0; t < 4; ++t)
        cf[t] = wmma4(a, load_b_frag(W_out, 64, k0, t * 16, lane), cf[t]);
    }
    __syncthreads();  // SF fully consumed; OUT may now overwrite it
#pragma unroll
    for (int t = 0; t < 4; ++t)
#pragma unroll
      for (int v = 0; v < 8; ++v) {
        int m = v + 8 * half;
        OUT[m * 130 + 64 + t * 16 + n] = cf[t][v];
      }
    // red rows -> OUT[:, 0:64]
#pragma unroll
    for (int it = 0; it < 8; ++it) {
      int r = it * 2 + half;  // 2 rows/iter, 16 lanes x float4 = 64 cols
      int c4 = (lane & 15) * 4;
      int gr = rw0 + r;
      gr = gr < M ? gr : M - 1;
      v4f v = *(const v4f*)(red + (size_t)gr * 64 + c4);
      float* d = OUT + r * 130 + c4;
      d[0] = v.x; d[1] = v.y; d[2] = v.z; d[3] = v.w;
    }
  }
  __syncthreads();

  // Phase 4: H = relu(OUT @ W_lin1) (16x64, K=128)
  {
    v8f ch[4] = {vzero, vzero, vzero, vzero};
#pragma unroll
    for (int k0 = 0; k0 < 128; k0 += 4) {
      v2f a = load_a_frag(OUT, 130, k0, lane);
#pragma unroll
      for (int t = 0; t < 4; ++t)
        ch[t] = wmma4(a, load_b_frag(W_lin1, 64, k0, t * 16, lane), ch[t]);
    }
    __syncthreads();  // ZH (Z) fully consumed in phase 2; reuse for H
#pragma unroll
    for (int t = 0; t < 4; ++t)
#pragma unroll
      for (int v = 0; v < 8; ++v) {
        int m = v + 8 * half;
        ZH[m * 66 + t * 16 + n] = fmaxf(ch[t][v], 0.0f);
      }
  }
  __syncthreads();

  // Phase 5: P = H @ W_lin2 + b_lin2 (16x64, K=64) -> proj (global)
  {
    v8f cp[4] = {vzero, vzero, vzero, vzero};
#pragma unroll
    for (int k0 = 0; k0 < 64; k0 += 4) {
      v2f a = load_a_frag(ZH, 66, k0, lane);
#pragma unroll
      for (int t = 0; t < 4; ++t)
        cp[t] = wmma4(a, load_b_frag(W_lin2, 64, k0, t * 16, lane), cp[t]);
    }
#pragma unroll
    for (int t = 0; t < 4; ++t) {
      float bv = b_lin2[t * 16 + n];
#pragma unroll
      for (int v = 0; v < 8; ++v) {
        int m = v + 8 * half;
        int gr = rw0 + m;
        if (gr < M) proj[(size_t)gr * 64 + t * 16 + n] = cp[t][v] + bv;
      }
    }
  }
}

// ---------------------------------------------------------------------------
// Host-side orchestration
// ---------------------------------------------------------------------------
extern "C" void kernel_launch(void* const* d_in, const int* in_sizes, int n_in,
                              void* d_out, int out_size, void* d_ws,
                              size_t ws_size, hipStream_t stream) {
  // dict order from setup_inputs():
  const int* inv0 = (const int*)d_in[0];
  const int* inv1 = (const int*)d_in[2];
  const int* inv2 = (const int*)d_in[4];
  const int* inv3 = (const int*)d_in[6];
  const int* inv_pts = (const int*)d_in[8];
  const int* inv_out = (const int*)d_in[9];
  const float* input_data = (const float*)d_in[11];
  const float* W_red = (const float*)d_in[12];
  const float* b_red = (const float*)d_in[13];
  const float* W_att = (const float*)d_in[14];
  const float* b_att = (const float*)d_in[15];
  const float* W_fcs = (const float*)d_in[16];
  const float* b_fcs = (const float*)d_in[17];
  const float* W_fc = (const float*)d_in[18];
  const float* W_out = (const float*)d_in[19];
  const float* W_lin1 = (const float*)d_in[20];
  const float* W_lin2 = (const float*)d_in[21];
  const float* b_lin2 = (const float*)d_in[22];

  const int NV = in_sizes[0];  // N_VOX (inv0 has one entry per voxel)
  const int NP = in_sizes[8];  // N_PTS

  // Workspace layout (floats). n_j only exists on device, so per-scale
  // buffers use the NV upper bound; unused rows compute zeros never gathered.
  float* red = (float*)d_ws;                   // NV*64
  float* ssum = red + (size_t)NV * 64;         // NV*64, later aliased by proj
  float* proj = ssum;
  float* cnt = ssum + (size_t)NV * 64;         // NV (directly follows ssum)
  float* att[4];
  att[0] = cnt + NV;                           // 4 x NV*32
  att[1] = att[0] + (size_t)NV * 32;
  att[2] = att[1] + (size_t)NV * 32;
  att[3] = att[2] + (size_t)NV * 32;
  const int* invs[4] = {inv0, inv1, inv2, inv3};

  const int gemm_blocks = (NV + 127) / 128;

  // 1) red = relu(X @ W_red + b_red)
  gemm_wmma_kernel<4, 64><<<gemm_blocks, 256, 0, stream>>>(
      input_data, nullptr, W_red, b_red, red, NV, 1);

  // 2) per-scale segment-mean + att GEMM
  for (int j = 0; j < 4; ++j) {
    long nz = (long)NV * 65;  // ssum + cnt (contiguous)
    fill_kernel<<<(int)((nz + 255) / 256), 256, 0, stream>>>(ssum, nz, 0.0f);
    segsum_kernel<<<(NV * 16 + 255) / 256, 256, 0, stream>>>(red, invs[j],
                                                             ssum, cnt, NV);
    gemm_wmma_kernel<2, 64><<<gemm_blocks, 256, 0, stream>>>(
        ssum, cnt, W_att + (size_t)j * 64 * 32, b_att + j * 32, att[j], NV, 1);
  }

  // 3) fused attention + MLP tail -> proj
  fuse_kernel<<<(NV + 63) / 64, 128, 0, stream>>>(
      red, att[0], att[1], att[2], att[3], inv0, inv1, inv2, inv3, W_fc, W_fcs,
      b_fcs, W_out, W_lin1, W_lin2, b_lin2, proj, NV);

  // 4) segment_max over points into d_out (out_size = n_out*64, host-known)
  fill_kernel<<<(out_size + 255) / 256, 256, 0, stream>>>((float*)d_out,
                                                          out_size,
                                                          -__builtin_inff());
  scatter_max_kernel<<<(NP * 16 + 255) / 256, 256, 0, stream>>>(
      proj, inv_pts, inv_out, (float*)d_out, NP);
}